// TaskAlignedAssigner_20907900797508
// MI455X (gfx1250) — compile-verified
//
#include <hip/hip_runtime.h>
#include <hip/hip_bf16.h>
#include <float.h>

// Problem constants (match reference setup_inputs)
constexpr int B_    = 16;
constexpr int NA_   = 8400;
constexpr int NMAX_ = 64;
constexpr int NC_   = 80;
constexpr int TOPK_ = 13;
constexpr float EPS_     = 1e-9f;
constexpr float IOU_EPS_ = 1e-7f;
constexpr float FOUR_OVER_PI2 = 0.40528473456935108577f; // 4/pi^2

typedef __attribute__((ext_vector_type(2))) float v2f;
typedef __attribute__((ext_vector_type(8))) float v8f;

// ---------------------------------------------------------------------------
// Kernel 1: bbox_scores[b,g,a] = pd_scores[b,a, gt_labels[b,g]]
// as a one-hot GEMM on the WMMA pipe:
//   C(16 anchors x 16 gts) += A(16 anchors x 4 classes) * B(4 classes x 16 gts)
// iterated over K = 80 classes in 20 steps of V_WMMA_F32_16X16X4_F32.
// A layout (ISA 7.12.2, 32-bit A 16x4): lane L holds row M=L&15;
//   lanes 0-15: VGPR0=K0,VGPR1=K1; lanes 16-31: VGPR0=K2,VGPR1=K3.
// B layout mirrors (lanes 0-15 hold K=0..1, 16-31 hold K=2..3, N=L&15).
// C layout: lane L, VGPR r -> M = r + (L>=16)*8, N = L&15.
// ---------------------------------------------------------------------------
__global__ __launch_bounds__(32)
void k_scores_wmma(const float* __restrict__ pd_scores,
                   const int* __restrict__ gt_labels,
                   float* __restrict__ bbox_scores) {
  const int a0   = blockIdx.x * 16;
  const int g0   = blockIdx.y * 16;
  const int b    = blockIdx.z;
  const int lane = threadIdx.x;
  const int half = lane >> 4;   // 0 or 1
  const int l15  = lane & 15;

  const int gl = gt_labels[b * NMAX_ + g0 + l15];           // this lane's gt column label
  const float* arow = pd_scores + ((size_t)b * NA_ + (a0 + l15)) * NC_;

  v8f c = {};
  #pragma unroll 4
  for (int kk = 0; kk < NC_; kk += 4) {
    const int kb = kk + half * 2;
    v2f av; av.x = arow[kb]; av.y = arow[kb + 1];
    v2f bv; bv.x = (gl == kb) ? 1.0f : 0.0f;
            bv.y = (gl == kb + 1) ? 1.0f : 0.0f;
    c = __builtin_amdgcn_wmma_f32_16x16x4_f32(false, av, false, bv,
                                              (short)0, c, false, false);
  }
  // Lane writes 8 consecutive anchors for its gt column.
  float* out = bbox_scores + ((size_t)b * NMAX_ + (g0 + l15)) * NA_ + a0 + half * 8;
  #pragma unroll
  for (int r = 0; r < 8; ++r) out[r] = c[r];
}

// ---------------------------------------------------------------------------
// Kernel 2: per-pair CIoU + align metric (score^1 * overlap^6), gated by
// mask_in_gts * mask_gt. grid: (ceil(NA/256), B*NMAX)
// ---------------------------------------------------------------------------
__global__ __launch_bounds__(256)
void k_metrics(const float* __restrict__ pd_bboxes,
               const float* __restrict__ anc_points,
               const float* __restrict__ gt_bboxes,
               const float* __restrict__ mask_gt,
               const float* __restrict__ bbox_scores,
               float* __restrict__ align_m,
               float* __restrict__ overlaps) {
  const int a = blockIdx.x * 256 + threadIdx.x;
  if (a >= NA_) return;
  const int row = blockIdx.y;          // b*NMAX + g
  const int b = row >> 6;
  const size_t off = (size_t)row * NA_ + a;

  const float mg = mask_gt[row];
  const float g_x1 = gt_bboxes[row * 4 + 0], g_y1 = gt_bboxes[row * 4 + 1];
  const float g_x2 = gt_bboxes[row * 4 + 2], g_y2 = gt_bboxes[row * 4 + 3];
  const float* pb = pd_bboxes + ((size_t)b * NA_ + a) * 4;
  const float p_x1 = pb[0], p_y1 = pb[1], p_x2 = pb[2], p_y2 = pb[3];
  const float ax = anc_points[a * 2], ay = anc_points[a * 2 + 1];

  const float din = fminf(fminf(ax - g_x1, ay - g_y1), fminf(g_x2 - ax, g_y2 - ay));
  const bool valid = (din > EPS_) && (mg > 0.0f);

  float align = 0.0f, ovl = 0.0f;
  if (valid) {
    const float w1 = g_x2 - g_x1, h1 = g_y2 - g_y1;
    const float w2 = p_x2 - p_x1, h2 = p_y2 - p_y1;
    const float iw = fmaxf(fminf(g_x2, p_x2) - fmaxf(g_x1, p_x1), 0.0f);
    const float ih = fmaxf(fminf(g_y2, p_y2) - fmaxf(g_y1, p_y1), 0.0f);
    const float inter = iw * ih;
    const float uni = w1 * h1 + w2 * h2 - inter + IOU_EPS_;
    const float iou = inter / uni;
    const float cw = fmaxf(g_x2, p_x2) - fminf(g_x1, p_x1);
    const float ch = fmaxf(g_y2, p_y2) - fminf(g_y1, p_y1);
    const float c2 = cw * cw + ch * ch + IOU_EPS_;
    const float dx = (p_x1 + p_x2 - g_x1 - g_x2);
    const float dy = (p_y1 + p_y2 - g_y1 - g_y2);
    const float rho2 = (dx * dx + dy * dy) * 0.25f;
    const float dat = atanf(w2 / h2) - atanf(w1 / h1);
    const float v = FOUR_OVER_PI2 * dat * dat;
    const float alpha = v / (v - iou + (1.0f + IOU_EPS_));
    const float ciou = iou - (rho2 / c2 + v * alpha);
    ovl = fmaxf(ciou, 0.0f);
    const float sc = bbox_scores[off];
    const float o2 = ovl * ovl;
    align = sc * o2 * o2 * o2;          // ALPHA=1, BETA=6
  }
  align_m[off] = align;
  overlaps[off] = ovl;
}

// ---------------------------------------------------------------------------
// Kernel 3: top-13 per (b,g) row. Stage the 8400-float row into LDS with
// CDNA5 async global->LDS copies (ASYNCcnt), then 13 block-argmax rounds
// with lowest-index tie-break (matches jax.lax.top_k). Emits mask_pos bytes
// = topk membership * mask_in_gts * (mask_gt>0).
// ---------------------------------------------------------------------------
__global__ __launch_bounds__(256)
void k_topk(const float* __restrict__ align_m,
            const float* __restrict__ anc_points,
            const float* __restrict__ gt_bboxes,
            const float* __restrict__ mask_gt,
            unsigned char* __restrict__ mask_pos) {
  const int row = blockIdx.x;          // b*NMAX + g
  const int tid = threadIdx.x;
  __shared__ float vals[NA_];          // 33.6 KB of the 320 KB WGP LDS
  __shared__ float rv[256];
  __shared__ int   ri[256];
  __shared__ int   sel[TOPK_];

  unsigned char* mrow = mask_pos + (size_t)row * NA_;
  const float mg = mask_gt[row];       // uniform per block
  if (mg <= 0.0f) {                    // whole row suppressed
    for (int i = tid; i < NA_; i += 256) mrow[i] = 0;
    return;
  }

  const float* arow = align_m + (size_t)row * NA_;
  // Async stage: per-lane global->LDS DMA, tracked by ASYNCcnt.
  for (int i = tid; i < NA_; i += 256) {
    const float* gp = arow + i;
    unsigned lds = (unsigned)(size_t)&vals[i];   // low 32 bits = LDS offset
    asm volatile("global_load_async_to_lds_b32 %0, %1, off"
                 :: "v"(lds), "v"(gp) : "memory");
  }
  asm volatile("s_wait_asynccnt 0x0" ::: "memory");
  __syncthreads();

  for (int t = 0; t < TOPK_; ++t) {
    float bv = -1.0f; int bi = 0;      // all live values >= 0; removed = -2
    for (int i = tid; i < NA_; i += 256) {
      const float v = vals[i];
      if (v > bv) { bv = v; bi = i; }  // ascending i + strict > => lowest index
    }
    rv[tid] = bv; ri[tid] = bi;
    __syncthreads();
    for (int s = 128; s > 0; s >>= 1) {
      if (tid < s) {
        const float ov = rv[tid + s]; const int oi = ri[tid + s];
        if (ov > rv[tid] || (ov == rv[tid] && oi < ri[tid])) { rv[tid] = ov; ri[tid] = oi; }
      }
      __syncthreads();
    }
    if (tid == 0) { sel[t] = ri[0]; vals[ri[0]] = -2.0f; }
    __syncthreads();
  }

  for (int i = tid; i < NA_; i += 256) mrow[i] = 0;
  __syncthreads();
  if (tid < TOPK_) {
    const int a = sel[tid];
    const float ax = anc_points[a * 2], ay = anc_points[a * 2 + 1];
    const float din = fminf(fminf(ax - gt_bboxes[row * 4 + 0],
                                  ay - gt_bboxes[row * 4 + 1]),
                            fminf(gt_bboxes[row * 4 + 2] - ax,
                                  gt_bboxes[row * 4 + 3] - ay));
    if (din > EPS_) mrow[a] = 1;       // mask_in_gts gate (mask_gt>0 already known)
  }
}

// ---------------------------------------------------------------------------
// Kernel 4: per-anchor conflict resolution + gather of label/bbox targets.
// ---------------------------------------------------------------------------
__global__ __launch_bounds__(256)
void k_resolve(const float* __restrict__ overlaps,
               unsigned char* __restrict__ mask_pos,
               const int* __restrict__ gt_labels,
               const float* __restrict__ gt_bboxes,
               float* __restrict__ out_labels,
               float* __restrict__ out_bboxes,
               float* __restrict__ out_fg,
               float* __restrict__ out_tgt) {
  const int idx = blockIdx.x * 256 + threadIdx.x;   // b*NA + a
  if (idx >= B_ * NA_) return;
  const int b = idx / NA_;
  const int a = idx - b * NA_;

  int fg = 0, firstj = -1, bj = 0;
  float bov = -FLT_MAX;
  for (int j = 0; j < NMAX_; ++j) {
    const size_t off = ((size_t)(b * NMAX_ + j)) * NA_ + a;
    const int m = mask_pos[off];
    fg += m;
    if (m && firstj < 0) firstj = j;
    const float ov = overlaps[off];
    if (ov > bov) { bov = ov; bj = j; }              // argmax, lowest index ties
  }
  int tgt, fgm;
  if (fg > 1) {                                      // multi-assigned anchor
    tgt = bj; fgm = 1;
    for (int j = 0; j < NMAX_; ++j)
      mask_pos[((size_t)(b * NMAX_ + j)) * NA_ + a] = (j == bj) ? 1 : 0;
  } else {
    tgt = (firstj >= 0) ? firstj : 0;
    fgm = fg;
  }
  int lbl = gt_labels[b * NMAX_ + tgt]; if (lbl < 0) lbl = 0;
  out_labels[idx] = (float)lbl;
  out_fg[idx]     = (float)fgm;
  out_tgt[idx]    = (float)tgt;
  const float* gb = gt_bboxes + ((size_t)(b * NMAX_ + tgt)) * 4;
  float* ob = out_bboxes + (size_t)idx * 4;
  ob[0] = gb[0]; ob[1] = gb[1]; ob[2] = gb[2]; ob[3] = gb[3];
}

// ---------------------------------------------------------------------------
// Kernel 5: per-row maxima of masked align / overlaps.
// ---------------------------------------------------------------------------
__global__ __launch_bounds__(256)
void k_rowmax(const float* __restrict__ align_m,
              const float* __restrict__ overlaps,
              const unsigned char* __restrict__ mask_pos,
              float* __restrict__ pos_align,
              float* __restrict__ pos_ovl) {
  const int row = blockIdx.x;
  const int tid = threadIdx.x;
  __shared__ float ra[256], ro[256];
  float ma = 0.0f, mo = 0.0f;
  const size_t base = (size_t)row * NA_;
  for (int i = tid; i < NA_; i += 256) {
    if (mask_pos[base + i]) {
      ma = fmaxf(ma, align_m[base + i]);
      mo = fmaxf(mo, overlaps[base + i]);
    }
  }
  ra[tid] = ma; ro[tid] = mo;
  __syncthreads();
  for (int s = 128; s > 0; s >>= 1) {
    if (tid < s) {
      ra[tid] = fmaxf(ra[tid], ra[tid + s]);
      ro[tid] = fmaxf(ro[tid], ro[tid + s]);
    }
    __syncthreads();
  }
  if (tid == 0) { pos_align[row] = ra[0]; pos_ovl[row] = ro[0]; }
}

// ---------------------------------------------------------------------------
// Kernel 6: normalization + one-hot target_scores.
// ---------------------------------------------------------------------------
__global__ __launch_bounds__(256)
void k_scores_out(const float* __restrict__ align_m,
                  const unsigned char* __restrict__ mask_pos,
                  const float* __restrict__ pos_align,
                  const float* __restrict__ pos_ovl,
                  const float* __restrict__ out_labels,
                  const float* __restrict__ out_fg,
                  float* __restrict__ out_scores) {
  const int idx = blockIdx.x * 256 + threadIdx.x;   // b*NA + a
  if (idx >= B_ * NA_) return;
  const int b = idx / NA_;
  const int a = idx - b * NA_;
  float norm = 0.0f;
  for (int j = 0; j < NMAX_; ++j) {
    const int row = b * NMAX_ + j;
    const size_t off = (size_t)row * NA_ + a;
    if (mask_pos[off]) {
      const float v = align_m[off] * pos_ovl[row] / (pos_align[row] + EPS_);
      norm = fmaxf(norm, v);
    }
  }
  const int lab = (int)out_labels[idx];
  const float val = (out_fg[idx] > 0.0f) ? norm : 0.0f;
  float* srow = out_scores + (size_t)idx * NC_;
  #pragma unroll 8
  for (int c = 0; c < NC_; ++c) srow[c] = (c == lab) ? val : 0.0f;
}

// ---------------------------------------------------------------------------
extern "C" void kernel_launch(void* const* d_in, const int* in_sizes, int n_in,
                              void* d_out, int out_size, void* d_ws, size_t ws_size,
                              hipStream_t stream) {
  const float* pd_scores  = (const float*)d_in[0];
  const float* pd_bboxes  = (const float*)d_in[1];
  const float* anc_points = (const float*)d_in[2];
  /* d_in[3] = rfields: unused by reference */
  const int*   gt_labels  = (const int*)d_in[4];
  const float* gt_bboxes  = (const float*)d_in[5];
  const float* mask_gt    = (const float*)d_in[6];

  const size_t PAIRS = (size_t)B_ * NMAX_ * NA_;     // 8,601,600
  char* ws = (char*)d_ws;
  float* bbox_scores = (float*)ws;               ws += PAIRS * 4;
  float* align_m     = (float*)ws;               ws += PAIRS * 4;
  float* overlaps    = (float*)ws;               ws += PAIRS * 4;
  unsigned char* mask_pos = (unsigned char*)ws;  ws += PAIRS;       // multiple of 256
  float* pos_align   = (float*)ws;               ws += B_ * NMAX_ * 4;
  float* pos_ovl     = (float*)ws;               ws += B_ * NMAX_ * 4;

  float* out_labels = (float*)d_out;                             // B*NA
  float* out_bboxes = out_labels + (size_t)B_ * NA_;             // B*NA*4
  float* out_scores = out_bboxes + (size_t)B_ * NA_ * 4;         // B*NA*NC
  float* out_fg     = out_scores + (size_t)B_ * NA_ * NC_;       // B*NA
  float* out_tgt    = out_fg + (size_t)B_ * NA_;                 // B*NA

  dim3 g1(NA_ / 16, NMAX_ / 16, B_);                 // 525 x 4 x 16 WMMA tiles
  k_scores_wmma<<<g1, 32, 0, stream>>>(pd_scores, gt_labels, bbox_scores);

  dim3 g2((NA_ + 255) / 256, B_ * NMAX_);
  k_metrics<<<g2, 256, 0, stream>>>(pd_bboxes, anc_points, gt_bboxes, mask_gt,
                                    bbox_scores, align_m, overlaps);

  k_topk<<<B_ * NMAX_, 256, 0, stream>>>(align_m, anc_points, gt_bboxes,
                                         mask_gt, mask_pos);

  const int NT = B_ * NA_;
  k_resolve<<<(NT + 255) / 256, 256, 0, stream>>>(overlaps, mask_pos, gt_labels,
                                                  gt_bboxes, out_labels, out_bboxes,
                                                  out_fg, out_tgt);

  k_rowmax<<<B_ * NMAX_, 256, 0, stream>>>(align_m, overlaps, mask_pos,
                                           pos_align, pos_ovl);

  k_scores_out<<<(NT + 255) / 256, 256, 0, stream>>>(align_m, mask_pos, pos_align,
                                                     pos_ovl, out_labels, out_fg,
                                                     out_scores);
}